// MLPLoraSubspace_36747740184734
// MI455X (gfx1250) — compile-verified
//
#include <hip/hip_runtime.h>
#include <math.h>

// ---------------------------------------------------------------------------
// MLPLoraSubspace on MI455X (gfx1250, wave32).
//
// Rank-1 weight synthesis means x @ W.T == (x @ Bv) outer A, and the BN batch
// statistics of a rank-1 matrix reduce to scalar stats of s = x @ Bv
// (linear_bias cancels inside BN). Mandatory HBM traffic: 268 MB (x) +
// 16.8 MB (out) ~= 285 MB -> ~12.3 us at 23.3 TB/s. The contraction itself is
// executed on the matrix pipe with V_WMMA_F32_16X16X4_F32.
// ---------------------------------------------------------------------------

#define BN_EPS    1e-5f
#define NEG_SLOPE 0.2f
#define BATCH 16384
#define DIN   4096
#define DOUT  256
#define DRANK 20

typedef float v2f __attribute__((ext_vector_type(2)));
typedef float v8f __attribute__((ext_vector_type(8)));

// ws layout (float offsets)
#define WS_A      0        // 256
#define WS_BV     256      // 4096
#define WS_SPART  4352     // 2 * 16384 (split-K partials)
#define WS_S      37120    // 16384
#define WS_ACCUM  53504    // 2 (sum, sumsq)  -> total ~209 KB

// --- K0: synthesize A and Bv, zero accumulators (every call, deterministic) --
__global__ void prep_kernel(const float* __restrict__ alphas,
                            const float* __restrict__ cA,   // (d, 256)
                            const float* __restrict__ cB,   // (4096, d)
                            float* __restrict__ ws) {
  const int t = threadIdx.x;
  float a = 0.f;
#pragma unroll
  for (int d = 0; d < DRANK; ++d) a += alphas[d] * cA[d * DOUT + t];
  ws[WS_A + t] = a;

  for (int k = t; k < DIN; k += 256) {
    float b = 0.f;
#pragma unroll
    for (int d = 0; d < DRANK; ++d) b += alphas[d] * cB[k * DRANK + d];
    ws[WS_BV + k] = b;
  }
  if (t < 2) ws[WS_ACCUM + t] = 0.f;
}

// --- K1: s_partial = x . Bv via V_WMMA_F32_16X16X4_F32 ----------------------
// One wave per (16-row tile, K-half). A-frag: 16x4 f32 native layout
// (lane = M, VGPR0/1 = contiguous K pair, +2 for lanes 16..31) -> one b64
// load per lane per K-step; every x element is read exactly once.
// B-frag: Bv[k..k+3] broadcast into all 16 N columns, so D[m][n] = s_m for
// every n; lane 0 / lane 16 (N==0) write the 16 row sums.
__global__ void gemv_wmma_kernel(const float* __restrict__ x,
                                 const float* __restrict__ Bv,
                                 float* __restrict__ s_part) {
  const int wave  = threadIdx.x >> 5;                 // 4 waves / block
  const int lane  = threadIdx.x & 31;
  const int tile  = blockIdx.x * 4 + wave;            // 0..2047
  const int kHalf = tile & 1;                         // split-K = 2
  const int row0  = (tile >> 1) * 16;                 // 16-row tile
  const int kBase = kHalf * (DIN / 2);

  const int m  = lane & 15;
  const int hi = lane >> 4;                           // 0: K0/K1, 1: K2/K3
  const float* xrow = x + (size_t)(row0 + m) * DIN + kBase + 2 * hi;
  const float* bvp  = Bv + kBase;

  v8f c = {};
#pragma unroll 4
  for (int k = 0; k < DIN / 2; k += 4) {
    v2f a = *(const v2f*)(xrow + k);
    float4 bv = *(const float4*)(bvp + k);            // wave-uniform 16B load
    v2f b;
    b.x = hi ? bv.z : bv.x;
    b.y = hi ? bv.w : bv.y;
    c = __builtin_amdgcn_wmma_f32_16x16x4_f32(
        /*neg_a=*/false, a, /*neg_b=*/false, b,
        /*c_mod=*/(short)0, c, /*reuse_a=*/false, /*reuse_b=*/false);
  }

  // C/D layout: lanes 0-15 hold M=0..7 in c[0..7]; lanes 16-31 hold M=8..15.
  if (m == 0) {
    float* out = s_part + (size_t)kHalf * BATCH + row0 + hi * 8;
#pragma unroll
    for (int i = 0; i < 8; ++i) out[i] = c[i];
  }
}

// --- K2: s = p0 + p1; accumulate sum(s), sum(s^2) ---------------------------
__global__ void stats_kernel(float* __restrict__ ws) {
  const float* p0 = ws + WS_SPART;
  const float* p1 = ws + WS_SPART + BATCH;
  float* s     = ws + WS_S;
  float* accum = ws + WS_ACCUM;

  __shared__ float r0[256];
  __shared__ float r1[256];

  const int i = blockIdx.x * 256 + threadIdx.x;
  const float v = p0[i] + p1[i];
  s[i] = v;
  r0[threadIdx.x] = v;
  r1[threadIdx.x] = v * v;
  __syncthreads();
  for (int off = 128; off > 0; off >>= 1) {
    if (threadIdx.x < off) {
      r0[threadIdx.x] += r0[threadIdx.x + off];
      r1[threadIdx.x] += r1[threadIdx.x + off];
    }
    __syncthreads();
  }
  if (threadIdx.x == 0) {
    atomicAdd(&accum[0], r0[0]);
    atomicAdd(&accum[1], r1[0]);
  }
}

// --- K3: fused BN + LeakyReLU rank-1 expansion, float4 stores ---------------
__device__ __forceinline__ float bn_act(float a, float g, float b,
                                        float sv, float var) {
  const float cj = a * g * rsqrtf(var * a * a + BN_EPS);
  const float v = sv * cj + b;
  return v >= 0.f ? v : NEG_SLOPE * v;
}

__global__ void epilogue_kernel(const float* __restrict__ ws,
                                const float* __restrict__ gamma,
                                const float* __restrict__ beta,
                                float* __restrict__ out) {
  const float* A     = ws + WS_A;
  const float* s     = ws + WS_S;
  const float* accum = ws + WS_ACCUM;

  const float inv_n = 1.0f / (float)BATCH;
  const float mean  = accum[0] * inv_n;
  const float var   = fmaxf(accum[1] * inv_n - mean * mean, 0.f);

  const int t   = threadIdx.x;
  const int j   = (t & 63) * 4;                       // 64 thr span one row
  const int row = blockIdx.x * 4 + (t >> 6);

  const float sv = s[row] - mean;
  const float4 a4 = *(const float4*)(A + j);
  const float4 g4 = *(const float4*)(gamma + j);
  const float4 b4 = *(const float4*)(beta + j);

  float4 o;
  o.x = bn_act(a4.x, g4.x, b4.x, sv, var);
  o.y = bn_act(a4.y, g4.y, b4.y, sv, var);
  o.z = bn_act(a4.z, g4.z, b4.z, sv, var);
  o.w = bn_act(a4.w, g4.w, b4.w, sv, var);
  *(float4*)(out + (size_t)row * DOUT + j) = o;
}

extern "C" void kernel_launch(void* const* d_in, const int* in_sizes, int n_in,
                              void* d_out, int out_size, void* d_ws, size_t ws_size,
                              hipStream_t stream) {
  const float* x      = (const float*)d_in[0];  // (16384, 4096)
  const float* alphas = (const float*)d_in[1];  // (20, 1)
  const float* cA     = (const float*)d_in[2];  // (20, 256)
  const float* cB     = (const float*)d_in[3];  // (4096, 20)
  // d_in[4] = linear_bias: cancels exactly inside BatchNorm (mean shift only).
  const float* gamma  = (const float*)d_in[5];
  const float* beta   = (const float*)d_in[6];
  float* ws  = (float*)d_ws;
  float* out = (float*)d_out;

  prep_kernel<<<1, 256, 0, stream>>>(alphas, cA, cB, ws);
  gemv_wmma_kernel<<<512, 128, 0, stream>>>(x, ws + WS_BV, ws + WS_SPART);
  stats_kernel<<<64, 256, 0, stream>>>(ws);
  epilogue_kernel<<<4096, 256, 0, stream>>>(ws, gamma, beta, out);
}